// PointerNetwork_18021682774829
// MI455X (gfx1250) — compile-verified
//
#include <hip/hip_runtime.h>
#include <math.h>

typedef __attribute__((ext_vector_type(2))) float v2f;
typedef __attribute__((ext_vector_type(8))) float v8f;

#define Hdim 256
#define ENC 1024
#define DEC 256
#define BATCH 4

// Guaranteed hardware tanh on gfx1250: prefer the clang builtin (lets the
// scheduler handle the TRANS-op hazard), else emit v_tanh_f32 directly.
// The trailing v_nop satisfies the ISA rule that a TRANS result needs one
// independent op before use (compiler cannot see through asm to schedule it).
__device__ __forceinline__ float gpu_tanh(float x) {
#if __has_builtin(__builtin_amdgcn_tanhf)
    return __builtin_amdgcn_tanhf(x);
#elif __has_builtin(__builtin_amdgcn_tanh_f32)
    return __builtin_amdgcn_tanh_f32(x);
#else
    float r;
    asm volatile("v_tanh_f32 %0, %1\n\tv_nop" : "=v"(r) : "v"(x));
    return r;
#endif
}

// out[r][o] = sum_h x[r][h] * w[o][h]     x: rows x 256, w: 256 x 256 (row-major, (out,in))
// One wave (32 threads) per 16x16 output tile, K marched in steps of 4 via
// V_WMMA_F32_16X16X4_F32.
// A-layout (16x4 f32): lanes 0-15 -> M=lane, K = {k0, k0+1}; lanes 16-31 -> K = {k0+2, k0+3}.
// B-layout (4x16 f32): lanes 0-15 -> N=lane, K = {k0, k0+1}; lanes 16-31 -> K = {k0+2, k0+3}.
// C/D (16x16 f32): vgpr r -> row M = r + 8*(lane/16), N = lane%16.
__global__ __launch_bounds__(32) void gemm_xwT_wmma(const float* __restrict__ x,
                                                    const float* __restrict__ w,
                                                    float* __restrict__ out) {
    const int lane  = threadIdx.x;          // 0..31, full wave32
    const int mtile = blockIdx.x;
    const int ntile = blockIdx.y;           // 0..15 (H/16)
    const int l15   = lane & 15;
    const int koff  = (lane >> 4) << 1;     // 0 or 2

    const float2* xrow = reinterpret_cast<const float2*>(x + (size_t)(mtile * 16 + l15) * Hdim);
    const float2* wrow = reinterpret_cast<const float2*>(w + (size_t)(ntile * 16 + l15) * Hdim);

    v8f c = {};
#pragma unroll 4
    for (int k0 = 0; k0 < Hdim; k0 += 4) {
        float2 af = xrow[(k0 + koff) >> 1];
        float2 bf = wrow[(k0 + koff) >> 1];
        v2f a; a.x = af.x; a.y = af.y;
        v2f b; b.x = bf.x; b.y = bf.y;
        // 8 args: (neg_a, A, neg_b, B, c_mod, C, reuse_a, reuse_b)
        c = __builtin_amdgcn_wmma_f32_16x16x4_f32(false, a, false, b, (short)0, c,
                                                  false, false);
    }

    const int nstore = ntile * 16 + l15;
    const int mbase  = mtile * 16 + ((lane >> 4) << 3);
#pragma unroll
    for (int r = 0; r < 8; ++r)
        out[(size_t)(mbase + r) * Hdim + nstore] = c[r];
}

// One block per (b,d). scores[e] = sum_h v[h]*tanh(dec[h] + enc[b,e,h]), then
// log_softmax over e in-block. 256 threads; thread t owns e = {t, t+256, t+512, t+768}.
__global__ __launch_bounds__(256) void attn_scores_lsm(const float* __restrict__ dec_t,
                                                       const float* __restrict__ enc_t,
                                                       const float* __restrict__ v,
                                                       float* __restrict__ out) {
    const int bd = blockIdx.x;      // b*DEC + d
    const int b  = bd / DEC;
    const int t  = threadIdx.x;     // 0..255

    __shared__ float sdec[Hdim];
    __shared__ float sv[Hdim];
    __shared__ float red[256];

    sdec[t] = dec_t[(size_t)bd * Hdim + t];
    sv[t]   = v[t];
    __syncthreads();

    const float4* sdec4 = reinterpret_cast<const float4*>(sdec);
    const float4* sv4   = reinterpret_cast<const float4*>(sv);

    float s[4];
#pragma unroll
    for (int j = 0; j < 4; ++j) {
        const int e = j * 256 + t;
        const float4* erow =
            reinterpret_cast<const float4*>(enc_t + ((size_t)b * ENC + e) * Hdim);
        float acc = 0.f;
        for (int i = 0; i < Hdim / 4; ++i) {
            float4 ev = erow[i];
            float4 dv = sdec4[i];   // same-address LDS read -> broadcast
            float4 vv = sv4[i];
            acc += vv.x * gpu_tanh(dv.x + ev.x);
            acc += vv.y * gpu_tanh(dv.y + ev.y);
            acc += vv.z * gpu_tanh(dv.z + ev.z);
            acc += vv.w * gpu_tanh(dv.w + ev.w);
        }
        s[j] = acc;
    }

    // block max
    float m = fmaxf(fmaxf(s[0], s[1]), fmaxf(s[2], s[3]));
    red[t] = m;
    __syncthreads();
    for (int str = 128; str >= 1; str >>= 1) {
        if (t < str) red[t] = fmaxf(red[t], red[t + str]);
        __syncthreads();
    }
    const float gmax = red[0];
    __syncthreads();

    // block sum of exp
    float lsum = 0.f;
#pragma unroll
    for (int j = 0; j < 4; ++j) lsum += __expf(s[j] - gmax);
    red[t] = lsum;
    __syncthreads();
    for (int str = 128; str >= 1; str >>= 1) {
        if (t < str) red[t] += red[t + str];
        __syncthreads();
    }
    const float lse = gmax + __logf(red[0]);

#pragma unroll
    for (int j = 0; j < 4; ++j)
        out[(size_t)bd * ENC + (size_t)j * 256 + t] = s[j] - lse;
}

extern "C" void kernel_launch(void* const* d_in, const int* in_sizes, int n_in,
                              void* d_out, int out_size, void* d_ws, size_t ws_size,
                              hipStream_t stream) {
    const float* x_dec = (const float*)d_in[0];   // (4, 256, 256)
    const float* x_enc = (const float*)d_in[1];   // (4, 1024, 256)
    const float* w1    = (const float*)d_in[2];   // (256, 256)
    const float* w2    = (const float*)d_in[3];   // (256, 256)
    const float* v     = (const float*)d_in[4];   // (256,)
    float* out = (float*)d_out;                   // (4, 256, 1024)

    float* enc_t = (float*)d_ws;                          // 4*1024*256 f32 = 4 MB
    float* dec_t = enc_t + (size_t)BATCH * ENC * Hdim;    // 4*256*256  f32 = 1 MB

    // enc_t = x_enc @ w1^T  : (BATCH*ENC) x H rows
    gemm_xwT_wmma<<<dim3(BATCH * ENC / 16, Hdim / 16), 32, 0, stream>>>(x_enc, w1, enc_t);
    // dec_t = x_dec @ w2^T  : (BATCH*DEC) x H rows
    gemm_xwT_wmma<<<dim3(BATCH * DEC / 16, Hdim / 16), 32, 0, stream>>>(x_dec, w2, dec_t);
    // scores + log_softmax
    attn_scores_lsm<<<BATCH * DEC, 256, 0, stream>>>(dec_t, enc_t, v, out);
}